// WPEV_loss_34969623724103
// MI455X (gfx1250) — compile-verified
//
#include <hip/hip_runtime.h>
#include <hip/hip_bf16.h>

// WPEV loss, fused single pass + WMMA-based wave reduction.
// Shapes are fixed by the reference: B=2, 6 channels, H=W=D=80.

typedef __attribute__((ext_vector_type(2))) float v2f;
typedef __attribute__((ext_vector_type(8))) float v8f;

#define HWD   512000   // 80*80*80
#define NVOX  1024000  // B * HWD, B = 2
#define NBLK  1024
#define NTHR  256

// Weighted principal eigenvector of symmetric 3x3:
//   ani = lmax - (l0+l1)/2 = 1.5*(lmax - trace/3),  out = ani * v(lmax)
__device__ __forceinline__ void wpev3(float a00, float a01, float a02,
                                      float a11, float a12, float a22,
                                      float& ox, float& oy, float& oz) {
  const float third = 1.0f / 3.0f;
  float q   = (a00 + a11 + a22) * third;
  float b00 = a00 - q, b11 = a11 - q, b22 = a22 - q;
  float p1  = a01 * a01 + a02 * a02 + a12 * a12;
  float p2  = b00 * b00 + b11 * b11 + b22 * b22 + 2.0f * p1;
  float p   = sqrtf(p2 * (1.0f / 6.0f));
  float invp = (p2 > 1e-24f) ? (1.0f / p) : 0.0f;   // isotropic -> detB=0 -> ani ~ 0
  float c00 = b00 * invp, c01 = a01 * invp, c02 = a02 * invp;
  float c11 = b11 * invp, c12 = a12 * invp, c22 = b22 * invp;
  float detB = c00 * (c11 * c22 - c12 * c12)
             - c01 * (c01 * c22 - c12 * c02)
             + c02 * (c01 * c12 - c11 * c02);
  float r    = fminf(1.0f, fmaxf(-1.0f, 0.5f * detB));
  float phi  = acosf(r) * third;
  float lmax = q + 2.0f * p * cosf(phi);
  float ani  = 1.5f * (lmax - q);

  // Eigenvector: largest cross product among rows of (A - lmax*I)
  float r0x = a00 - lmax, r0y = a01,        r0z = a02;
  float r1x = a01,        r1y = a11 - lmax, r1z = a12;
  float r2x = a02,        r2y = a12,        r2z = a22 - lmax;
  float e0x = r0y * r1z - r0z * r1y, e0y = r0z * r1x - r0x * r1z, e0z = r0x * r1y - r0y * r1x;
  float e1x = r0y * r2z - r0z * r2y, e1y = r0z * r2x - r0x * r2z, e1z = r0x * r2y - r0y * r2x;
  float e2x = r1y * r2z - r1z * r2y, e2y = r1z * r2x - r1x * r2z, e2z = r1x * r2y - r1y * r2x;
  float n0 = e0x * e0x + e0y * e0y + e0z * e0z;
  float n1 = e1x * e1x + e1y * e1y + e1z * e1z;
  float n2 = e2x * e2x + e2y * e2y + e2z * e2z;
  float vx = e0x, vy = e0y, vz = e0z, nn = n0;
  if (n1 > nn) { vx = e1x; vy = e1y; vz = e1z; nn = n1; }
  if (n2 > nn) { vx = e2x; vy = e2y; vz = e2z; nn = n2; }
  float inv = (nn > 1e-30f) ? rsqrtf(nn) : 0.0f;

  // Fixed sign convention (applied identically to input & target)
  float ax = fabsf(vx), ay = fabsf(vy), az = fabsf(vz);
  float mc = (ax >= ay) ? ((ax >= az) ? vx : vz) : ((ay >= az) ? vy : vz);
  float sgn = (mc < 0.0f) ? -1.0f : 1.0f;

  float sc = ani * inv * sgn;
  ox = vx * sc; oy = vy * sc; oz = vz * sc;
}

__global__ __launch_bounds__(NTHR)
void wpev_loss_main(const float* __restrict__ in, const float* __restrict__ tg,
                    const float* __restrict__ mask,
                    const float* __restrict__ gmean, const float* __restrict__ gstd,
                    float* __restrict__ partials) {
  float mean[6], stdv[6];
#pragma unroll
  for (int c = 0; c < 6; ++c) { mean[c] = gmean[c]; stdv[c] = gstd[c]; }

  float p_num = 0.0f, p_cnt = 0.0f;
  const int stride = NBLK * NTHR;
  for (int v = blockIdx.x * NTHR + threadIdx.x; v < NVOX; v += stride) {
    int b = v / HWD;
    int s = v - b * HWD;
    const float* ip = in + (size_t)b * 6 * HWD + s;
    const float* tp = tg + (size_t)b * 6 * HWD + s;
    float mval = mask[v];

    float si[6], st[6];
#pragma unroll
    for (int c = 0; c < 6; ++c) {
      float iv = ip[(size_t)c * HWD];
      float tv = tp[(size_t)c * HWD];
      si[c] = fmaf(iv, stdv[c], mean[c]) * mval;   // (x*std + mean) * m
      st[c] = fmaf(tv, stdv[c], mean[c]) * mval;
    }

    float ix, iy, iz, tx, ty, tz;
    wpev3(si[0], si[1], si[2], si[3], si[4], si[5], ix, iy, iz);
    wpev3(st[0], st[1], st[2], st[3], st[4], st[5], tx, ty, tz);

    float vox = (mval > 0.0f) ? 1.0f : 0.0f;
    p_num += (fabsf(ix - tx) + fabsf(iy - ty) + fabsf(iz - tz)) * vox;
    p_cnt += vox;
  }

  // ---- wave32 reduction via V_WMMA_F32_16X16X4_F32 with B = ones ----
  // A is 16x4 f32 (2 VGPRs/lane): lane L (<16) supplies A[L,0..1], lane L+16
  // supplies A[L,2..3]. With B = ones, D[m,n] = rowsum_m for every column n,
  // so (sum of this lane's 8 acc regs) + (partner lane ^16) = wave total.
  v2f aN; aN[0] = p_num; aN[1] = 0.0f;
  v2f aC; aC[0] = p_cnt; aC[1] = 0.0f;
  v2f bo; bo[0] = 1.0f;  bo[1] = 1.0f;
  v8f z = {};
  v8f dN = __builtin_amdgcn_wmma_f32_16x16x4_f32(false, aN, false, bo, (short)0, z, false, false);
  v8f dC = __builtin_amdgcn_wmma_f32_16x16x4_f32(false, aC, false, bo, (short)0, z, false, false);
  float sN = dN[0] + dN[1] + dN[2] + dN[3] + dN[4] + dN[5] + dN[6] + dN[7];
  float sC = dC[0] + dC[1] + dC[2] + dC[3] + dC[4] + dC[5] + dC[6] + dC[7];
  sN += __shfl_xor(sN, 16);
  sC += __shfl_xor(sC, 16);

  __shared__ float smN[NTHR / 32], smC[NTHR / 32];
  int wave = threadIdx.x >> 5;
  int lane = threadIdx.x & 31;
  if (lane == 0) { smN[wave] = sN; smC[wave] = sC; }
  __syncthreads();
  if (threadIdx.x == 0) {
    float tN = 0.0f, tC = 0.0f;
#pragma unroll
    for (int w = 0; w < NTHR / 32; ++w) { tN += smN[w]; tC += smC[w]; }
    partials[2 * blockIdx.x]     = tN;
    partials[2 * blockIdx.x + 1] = tC;
  }
}

__global__ __launch_bounds__(NTHR)
void wpev_loss_final(const float* __restrict__ partials, float* __restrict__ out) {
  __shared__ float sN[NTHR], sC[NTHR];
  float tN = 0.0f, tC = 0.0f;
  for (int i = threadIdx.x; i < NBLK; i += NTHR) {
    tN += partials[2 * i];
    tC += partials[2 * i + 1];
  }
  sN[threadIdx.x] = tN; sC[threadIdx.x] = tC;
  __syncthreads();
  for (int off = NTHR / 2; off > 0; off >>= 1) {
    if (threadIdx.x < off) {
      sN[threadIdx.x] += sN[threadIdx.x + off];
      sC[threadIdx.x] += sC[threadIdx.x + off];
    }
    __syncthreads();
  }
  if (threadIdx.x == 0) out[0] = sN[0] / (3.0f * sC[0]);
}

extern "C" void kernel_launch(void* const* d_in, const int* in_sizes, int n_in,
                              void* d_out, int out_size, void* d_ws, size_t ws_size,
                              hipStream_t stream) {
  const float* in    = (const float*)d_in[0];  // input_data (2,6,80,80,80)
  const float* tg    = (const float*)d_in[1];  // target     (2,6,80,80,80)
  const float* mask  = (const float*)d_in[2];  // mask       (2,1,1,80,80,80)
  const float* gmean = (const float*)d_in[3];  // gt_mean    (6)
  const float* gstd  = (const float*)d_in[4];  // gt_std     (6)
  // d_in[5] = ani_mask: unused by the reference
  float* partials = (float*)d_ws;              // NBLK * 2 floats (8 KB)
  float* out = (float*)d_out;

  wpev_loss_main<<<NBLK, NTHR, 0, stream>>>(in, tg, mask, gmean, gstd, partials);
  wpev_loss_final<<<1, NTHR, 0, stream>>>(partials, out);
}